// TripletLoss_24326694764949
// MI455X (gfx1250) — compile-verified
//
#include <hip/hip_runtime.h>
#include <hip/hip_bf16.h>

typedef __attribute__((ext_vector_type(2))) float v2f;
typedef __attribute__((ext_vector_type(8))) float v8f;

#define NPTS 4096
#define DIM  256
#define LDA  260            // LDS row stride in floats: 4*row banks -> conflict-free ds_load_b64
#define MARGIN_F 0.3f
#define NO_POS_F 1000000.0f
#define BIG_F    3.0e38f

// ---------------------------------------------------------------- init ws ---
__global__ void tl_init(float* accum) {
    accum[0] = 0.0f;   // loss sum
    accum[1] = 0.0f;   // valid count
}

// ------------------------------------------------------------- row norms ----
__global__ __launch_bounds__(256) void tl_rownorm(const float* __restrict__ X,
                                                  float* __restrict__ xx) {
    const int row  = blockIdx.x * 8 + (threadIdx.x >> 5);
    const int lane = threadIdx.x & 31;
    const float4* p = (const float4*)(X + (size_t)row * DIM);
    float s = 0.0f;
    #pragma unroll
    for (int i = 0; i < 2; ++i) {
        float4 v = p[lane + 32 * i];
        s += v.x * v.x + v.y * v.y + v.z * v.z + v.w * v.w;
    }
    s += __shfl_xor(s, 16, 32);
    s += __shfl_xor(s,  8, 32);
    s += __shfl_xor(s,  4, 32);
    s += __shfl_xor(s,  2, 32);
    s += __shfl_xor(s,  1, 32);
    if (lane == 0) xx[row] = s;
}

// ----------------------------------------------------------- main kernel ----
// One block = 16 rows of the distance matrix, 8 waves split the 256 column
// tiles. Gram tile via V_WMMA_F32_16X16X4_F32, fused distance + hardest
// positive/negative reduction. Never materializes the 4096^2 matrix.
__global__ __launch_bounds__(256) void tl_main(const float* __restrict__ X,
                                               const int*   __restrict__ labels,
                                               const float* __restrict__ xx,
                                               float* __restrict__ accum) {
    __shared__ float As[16 * LDA];
    __shared__ float xrow[16];
    __shared__ int   lrow[16];
    __shared__ float red_pos[8 * 16];
    __shared__ float red_neg[8 * 16];

    const int tid     = threadIdx.x;
    const int lane    = tid & 31;
    const int wv      = tid >> 5;
    const int rowBase = blockIdx.x * 16;

    // Stage the 16x256 fp32 A-stripe into LDS (16 threads per row, 4x float4).
    {
        const int r  = tid >> 4;
        const int c0 = (tid & 15) * 16;
        const float4* src = (const float4*)(X + (size_t)(rowBase + r) * DIM + c0);
        float4 f0 = src[0], f1 = src[1], f2 = src[2], f3 = src[3];
        float* dst = As + r * LDA + c0;
        *(float4*)(dst + 0)  = f0;
        *(float4*)(dst + 4)  = f1;
        *(float4*)(dst + 8)  = f2;
        *(float4*)(dst + 12) = f3;
    }
    if (tid < 16) {
        xrow[tid] = xx[rowBase + tid];
        lrow[tid] = labels[rowBase + tid];
    }
    __syncthreads();

    const int mi   = lane & 15;          // column within 16x16 tile
    const int koff = (lane >> 4) << 1;   // 0 or 2  (A/B f32 K sub-offset)
    const int hh   = (lane >> 4) << 3;   // 0 or 8  (C/D row-half offset)

    float posmax[8], negmin[8];
    #pragma unroll
    for (int r = 0; r < 8; ++r) { posmax[r] = -BIG_F; negmin[r] = BIG_F; }

    for (int n0 = wv * 16; n0 < NPTS; n0 += 128) {
        const float* Bp = X + (size_t)(n0 + mi) * DIM + koff;
        if (n0 + 128 < NPTS)  // uniform branch; hint next tile toward L2/L0
            __builtin_prefetch(X + (size_t)(n0 + 128 + mi) * DIM, 0, 1);

        const float xcol = xx[n0 + mi];
        const int   lcol = labels[n0 + mi];

        v8f acc = {0.f, 0.f, 0.f, 0.f, 0.f, 0.f, 0.f, 0.f};
        #pragma unroll 8
        for (int k = 0; k < DIM; k += 4) {
            v2f a = *(const v2f*)(As + mi * LDA + k + koff);   // ds_load_b64
            v2f b = *(const v2f*)(Bp + k);                     // global_load_b64 (L2-hot)
            acc = __builtin_amdgcn_wmma_f32_16x16x4_f32(
                false, a, false, b, (short)0, acc, false, false);
        }

        // Epilogue: d = sqrt(max(xx_m + xx_n - 2*g, eps)); masked max/min.
        #pragma unroll
        for (int r = 0; r < 8; ++r) {
            float d2 = xrow[hh + r] + xcol - 2.0f * acc[r];
            float d  = sqrtf(fmaxf(d2, 1e-12f));
            bool same = (lrow[hh + r] == lcol);
            posmax[r] = same ? fmaxf(posmax[r], d) : posmax[r];
            negmin[r] = same ? negmin[r] : fminf(negmin[r], d);
        }
    }

    // Reduce over the 16 columns held by each half-wave (masks stay in-half).
    #pragma unroll
    for (int r = 0; r < 8; ++r) {
        posmax[r] = fmaxf(posmax[r], __shfl_xor(posmax[r], 1, 32));
        posmax[r] = fmaxf(posmax[r], __shfl_xor(posmax[r], 2, 32));
        posmax[r] = fmaxf(posmax[r], __shfl_xor(posmax[r], 4, 32));
        posmax[r] = fmaxf(posmax[r], __shfl_xor(posmax[r], 8, 32));
        negmin[r] = fminf(negmin[r], __shfl_xor(negmin[r], 1, 32));
        negmin[r] = fminf(negmin[r], __shfl_xor(negmin[r], 2, 32));
        negmin[r] = fminf(negmin[r], __shfl_xor(negmin[r], 4, 32));
        negmin[r] = fminf(negmin[r], __shfl_xor(negmin[r], 8, 32));
    }
    if (mi == 0) {
        #pragma unroll
        for (int r = 0; r < 8; ++r) {
            red_pos[wv * 16 + hh + r] = posmax[r];
            red_neg[wv * 16 + hh + r] = negmin[r];
        }
    }
    __syncthreads();

    // Cross-wave reduce + per-row loss; one atomicAdd pair per block.
    if (tid < 16) {
        float p  = red_pos[tid];
        float ng = red_neg[tid];
        #pragma unroll
        for (int w = 1; w < 8; ++w) {
            p  = fmaxf(p,  red_pos[w * 16 + tid]);
            ng = fminf(ng, red_neg[w * 16 + tid]);
        }
        float ap = (p > -1e37f) ? p : NO_POS_F;   // diagonal => has_pos normally true
        float an = (ng < 1e37f) ? ng : 0.0f;
        bool valid = (ap < NO_POS_F) && (an > 0.0f);
        float per  = valid ? fmaxf(MARGIN_F + ap - an, 0.0f) : 0.0f;
        float cnt  = valid ? 1.0f : 0.0f;
        per += __shfl_xor(per, 1, 32); cnt += __shfl_xor(cnt, 1, 32);
        per += __shfl_xor(per, 2, 32); cnt += __shfl_xor(cnt, 2, 32);
        per += __shfl_xor(per, 4, 32); cnt += __shfl_xor(cnt, 4, 32);
        per += __shfl_xor(per, 8, 32); cnt += __shfl_xor(cnt, 8, 32);
        if (tid == 0) {
            atomicAdd(&accum[0], per);
            atomicAdd(&accum[1], cnt);
        }
    }
}

// -------------------------------------------------------------- finalize ----
__global__ void tl_finalize(const float* __restrict__ accum,
                            float* __restrict__ out) {
    float s = accum[0], c = accum[1];
    out[0] = (c > 0.0f) ? (s / fmaxf(c, 1.0f)) : 0.0f;
}

// ---------------------------------------------------------------- launch ----
extern "C" void kernel_launch(void* const* d_in, const int* in_sizes, int n_in,
                              void* d_out, int out_size, void* d_ws, size_t ws_size,
                              hipStream_t stream) {
    (void)in_sizes; (void)n_in; (void)out_size; (void)ws_size;
    const float* X      = (const float*)d_in[0];
    const int*   labels = (const int*)d_in[1];
    float*       out    = (float*)d_out;
    float*       ws     = (float*)d_ws;
    float* accum = ws;        // 2 floats: loss sum, valid count
    float* xx    = ws + 2;    // 4096 floats: row squared norms

    tl_init<<<1, 1, 0, stream>>>(accum);
    tl_rownorm<<<NPTS / 8, 256, 0, stream>>>(X, xx);
    tl_main<<<NPTS / 16, 256, 0, stream>>>(X, labels, xx, accum);
    tl_finalize<<<1, 1, 0, stream>>>(accum, out);
}